// DMPNN_19868518711533
// MI455X (gfx1250) — compile-verified
//
#include <hip/hip_runtime.h>

// ---------------------------------------------------------------------------
// D-MPNN on MI455X (gfx1250, wave32) using V_WMMA_F32_16X16X4_F32
// ---------------------------------------------------------------------------

typedef __attribute__((ext_vector_type(2))) float v2f;
typedef __attribute__((ext_vector_type(8))) float v8f;

#define N_NODES 50000
#define E_HALF  400000
#define E_TOT   800000
#define D_NODE  128
#define D_EDGE  32
#define UNITS   64
#define N_TILES (N_NODES / 16)   // 3125
#define E_TILES (E_TOT / 16)     // 50000
#define WAVES_PER_BLOCK 8

// One WMMA macro-step: A fragment (v2f, lane=M, half-wave=K pair) against the
// four 16-column tiles of the 64-wide weight matrix held transposed in LDS.
// wt points at WT + (lane&15)*STR + 2*half ; koff = 4*kk.
template <int STR>
__device__ __forceinline__ void wmma4(v2f a, const float* wt, int koff, v8f acc[4]) {
#pragma unroll
  for (int nn = 0; nn < 4; ++nn) {
    v2f b = *(const v2f*)(wt + nn * 16 * STR + koff);
    acc[nn] = __builtin_amdgcn_wmma_f32_16x16x4_f32(
        false, a, false, b, (short)0, acc[nn], false, false);
  }
}

// ---------------------------------------------------------------------------
// h0 = relu([node[src] || edge_feat] @ W1 + b1)      (E x 160) @ (160 x 64)
// ---------------------------------------------------------------------------
__global__ __launch_bounds__(256) void k_init(
    const float* __restrict__ nodef, const float* __restrict__ edgef,
    const int* __restrict__ src,
    const float* __restrict__ W1, const float* __restrict__ b1,
    float* __restrict__ h) {
  const int K = 160, STR = 162;               // padded stride (floats)
  __shared__ float WT[STR * 64];              // W1 transposed: WT[c*STR + k]
  for (int idx = threadIdx.x; idx < K * 64; idx += 256) {
    int k = idx >> 6, c = idx & 63;           // W1[k][c], row-major read
    WT[c * STR + k] = W1[idx];
  }
  __syncthreads();

  int lane = threadIdx.x & 31;
  int tile = blockIdx.x * WAVES_PER_BLOCK + (threadIdx.x >> 5);
  if (tile >= E_TILES) return;
  int e0 = tile * 16;
  int half = lane >> 4;
  int r = lane & 15;
  int e = e0 + r;
  int s = src[e];
  const float* arow = nodef + (long)s * D_NODE;
  const float* erow = edgef + (long)e * D_EDGE;
  const float* wt = WT + r * STR + 2 * half;

  v8f acc[4] = {v8f{}, v8f{}, v8f{}, v8f{}};
#pragma unroll 4
  for (int kk = 0; kk < 40; ++kk) {           // K = 160 = 40 * 4
    int col = kk * 4 + 2 * half;
    v2f a = (col < D_NODE) ? *(const v2f*)(arow + col)
                           : *(const v2f*)(erow + (col - D_NODE));
    wmma4<STR>(a, wt, kk * 4, acc);
  }
#pragma unroll
  for (int nn = 0; nn < 4; ++nn) {
    int col = nn * 16 + r;
    float bias = b1[col];
#pragma unroll
    for (int j = 0; j < 8; ++j) {
      int row = j + half * 8;
      float v = acc[nn][j] + bias;
      h[(long)(e0 + row) * UNITS + col] = v > 0.f ? v : 0.f;
    }
  }
}

// ---------------------------------------------------------------------------
// agg[dst[e]] += h[e]   (hardware f32 atomics in L2; agg stays L2-resident)
// ---------------------------------------------------------------------------
__global__ __launch_bounds__(256) void k_scatter(
    const float* __restrict__ h, const int* __restrict__ dst,
    float* __restrict__ agg) {
  int tid = blockIdx.x * 256 + threadIdx.x;   // E_TOT * 16 threads
  int e = tid >> 4;
  if (e >= E_TOT) return;
  int c = (tid & 15) * 4;
  const float4* hp = (const float4*)(h + (long)e * UNITS + c);
  float4 v = *hp;
  float* base = agg + (long)dst[e] * UNITS + c;
  unsafeAtomicAdd(base + 0, v.x);
  unsafeAtomicAdd(base + 1, v.y);
  unsafeAtomicAdd(base + 2, v.z);
  unsafeAtomicAdd(base + 3, v.w);
}

// ---------------------------------------------------------------------------
// h' = relu((agg[src] - valid*h[rev]) @ Wu + bu + h)   (E x 64) @ (64 x 64)
// rev(e) = (e + E_HALF) % E by construction; validated against src/dst.
// ---------------------------------------------------------------------------
__global__ __launch_bounds__(256) void k_step(
    const float* __restrict__ agg, const float* __restrict__ hin,
    const int* __restrict__ src, const int* __restrict__ dst,
    const float* __restrict__ Wu, const float* __restrict__ bu,
    float* __restrict__ hout) {
  const int STR = 66;
  __shared__ float WT[STR * 64];              // Wu transposed
  for (int idx = threadIdx.x; idx < 64 * 64; idx += 256) {
    int k = idx >> 6, c = idx & 63;
    WT[c * STR + k] = Wu[idx];
  }
  __syncthreads();

  int lane = threadIdx.x & 31;
  int tile = blockIdx.x * WAVES_PER_BLOCK + (threadIdx.x >> 5);
  if (tile >= E_TILES) return;
  int e0 = tile * 16;
  int half = lane >> 4;
  int r = lane & 15;
  int e = e0 + r;
  int s = src[e];
  int rv = (e < E_HALF) ? (e + E_HALF) : (e - E_HALF);
  // rev_valid check (branch-free mask so EXEC stays all-ones around WMMA)
  float m = ((src[rv] == dst[e]) && (dst[rv] == s)) ? 1.f : 0.f;
  const float* arow = agg + (long)s * UNITS;
  const float* hrev = hin + (long)rv * UNITS;
  const float* wt = WT + r * STR + 2 * half;

  v8f acc[4] = {v8f{}, v8f{}, v8f{}, v8f{}};
#pragma unroll 4
  for (int kk = 0; kk < 16; ++kk) {           // K = 64 = 16 * 4
    int col = kk * 4 + 2 * half;
    v2f a  = *(const v2f*)(arow + col);
    v2f hv = *(const v2f*)(hrev + col);
    a.x -= m * hv.x;
    a.y -= m * hv.y;
    wmma4<STR>(a, wt, kk * 4, acc);
  }
#pragma unroll
  for (int nn = 0; nn < 4; ++nn) {
    int col = nn * 16 + r;
    float bias = bu[col];
#pragma unroll
    for (int j = 0; j < 8; ++j) {
      int row = j + half * 8;
      long off = (long)(e0 + row) * UNITS + col;
      float v = acc[nn][j] + bias + hin[off];  // residual
      hout[off] = v > 0.f ? v : 0.f;
    }
  }
}

// ---------------------------------------------------------------------------
// out = relu([node_feature || agg] @ Wf + bf)   (N x 192) @ (192 x 64)
// ---------------------------------------------------------------------------
__global__ __launch_bounds__(256) void k_final(
    const float* __restrict__ nodef, const float* __restrict__ agg,
    const float* __restrict__ Wf, const float* __restrict__ bf,
    float* __restrict__ out) {
  const int K = 192, STR = 194;
  __shared__ float WT[STR * 64];              // Wf transposed (~48.5 KB)
  for (int idx = threadIdx.x; idx < K * 64; idx += 256) {
    int k = idx >> 6, c = idx & 63;
    WT[c * STR + k] = Wf[idx];
  }
  __syncthreads();

  int lane = threadIdx.x & 31;
  int tile = blockIdx.x * WAVES_PER_BLOCK + (threadIdx.x >> 5);
  if (tile >= N_TILES) return;                 // partial last block: whole waves drop out
  int n0 = tile * 16;
  int half = lane >> 4;
  int r = lane & 15;
  int n = n0 + r;
  const float* arow = nodef + (long)n * D_NODE;
  const float* grow = agg + (long)n * UNITS;
  const float* wt = WT + r * STR + 2 * half;

  v8f acc[4] = {v8f{}, v8f{}, v8f{}, v8f{}};
#pragma unroll 4
  for (int kk = 0; kk < 48; ++kk) {           // K = 192 = 48 * 4
    int col = kk * 4 + 2 * half;
    v2f a = (col < D_NODE) ? *(const v2f*)(arow + col)
                           : *(const v2f*)(grow + (col - D_NODE));
    wmma4<STR>(a, wt, kk * 4, acc);
  }
#pragma unroll
  for (int nn = 0; nn < 4; ++nn) {
    int col = nn * 16 + r;
    float bias = bf[col];
#pragma unroll
    for (int j = 0; j < 8; ++j) {
      int row = j + half * 8;
      float v = acc[nn][j] + bias;
      out[(long)(n0 + row) * UNITS + col] = v > 0.f ? v : 0.f;
    }
  }
}

// ---------------------------------------------------------------------------
extern "C" void kernel_launch(void* const* d_in, const int* in_sizes, int n_in,
                              void* d_out, int out_size, void* d_ws, size_t ws_size,
                              hipStream_t stream) {
  const float* nodef = (const float*)d_in[0];
  const float* edgef = (const float*)d_in[1];
  const int*   src   = (const int*)d_in[2];
  const int*   dst   = (const int*)d_in[3];
  const float* W1    = (const float*)d_in[4];
  const float* b1    = (const float*)d_in[5];
  const float* Wu    = (const float*)d_in[6];
  const float* bu    = (const float*)d_in[7];
  const float* Wf    = (const float*)d_in[8];
  const float* bf    = (const float*)d_in[9];
  float* out = (float*)d_out;

  char* ws = (char*)d_ws;
  float* agg = (float*)ws;                                   // N * 64 f32
  float* hA  = (float*)(ws + (size_t)N_NODES * UNITS * 4);   // E * 64 f32
  float* hB  = hA + (size_t)E_TOT * UNITS;                   // E * 64 f32

  const int edge_blocks    = E_TILES / WAVES_PER_BLOCK;                  // 6250
  const int scatter_blocks = (E_TOT * 16) / 256;                         // 50000
  const int node_blocks    = (N_TILES + WAVES_PER_BLOCK - 1) / WAVES_PER_BLOCK; // 391
  const size_t agg_bytes   = (size_t)N_NODES * UNITS * 4;

  k_init<<<edge_blocks, 256, 0, stream>>>(nodef, edgef, src, W1, b1, hA);

  float* hc = hA;
  float* hn = hB;
  for (int step = 0; step < 4; ++step) {
    hipMemsetAsync(agg, 0, agg_bytes, stream);
    k_scatter<<<scatter_blocks, 256, 0, stream>>>(hc, dst, agg);
    k_step<<<edge_blocks, 256, 0, stream>>>(agg, hc, src, dst, Wu, bu, hn);
    float* t = hc; hc = hn; hn = t;
  }

  hipMemsetAsync(agg, 0, agg_bytes, stream);
  k_scatter<<<scatter_blocks, 256, 0, stream>>>(hc, dst, agg);
  k_final<<<node_blocks, 256, 0, stream>>>(nodef, agg, Wf, bf, out);
}